// RaFlow_81260781240444
// MI455X (gfx1250) — compile-verified
//
#include <hip/hip_runtime.h>
#include <hip/hip_bf16.h>

// RaFlow SFR module for MI455X (gfx1250, wave32).
// Pipeline: gram_reduce<pass1> -> solve1 -> gram_reduce<pass2,mask> -> solve2 -> finalize.
// Moment reductions run on the matrix pipe via V_WMMA_F32_16X16X4_F32 Gram accumulation.

typedef __attribute__((ext_vector_type(2))) float v2f;
typedef __attribute__((ext_vector_type(8))) float v8f;

#define BATCH 1024
#define NPTS  4096
#define RIGID_THRES 0.15f
#define RIGID_PCS   0.25f

// LDS staging tile: [wave][16 rows][34] ; rows 7..15 stay zero (WMMA zero rows),
// pitch 34 => distinct banks per row AND 8B alignment for v2f (ds_load_b64) reads.
#define TPITCH 34
#define TROWS  16
#define TIDX(wv, row, p) ((wv) * (TROWS * TPITCH) + (row) * TPITCH + (p))

// ---------------------------------------------------------------------------
// Per-batch Gram-matrix reduction: G = X X^T with X rows =
//   [w*Ax, w*Ay, w*Az, w*Bx, w*By, w*Bz, w, 0, ..., 0]  (16 x N, B = A + output)
// G[i][3+j] = sum w*Ai*Bj ; G[i][6] = sum w*Ai ; G[3+j][6] = sum w*Bj ; G[6][6] = sum w
// One block (8 waves) per batch; each wave covers a 32-point window per tile,
// 8 x V_WMMA_F32_16X16X4_F32 per tile. Since G is symmetric, the 16x4 A-chunk
// and 4x16 B-chunk have identical per-lane VGPR layouts -> one operand feeds both.
// ---------------------------------------------------------------------------
template <bool PASS2>
__global__ __launch_bounds__(256) void gram_reduce(
    const float* __restrict__ pc, const float* __restrict__ outp,
    const float* __restrict__ feat, const float* __restrict__ interval,
    const float* __restrict__ RT1, float* __restrict__ G,
    float* __restrict__ maskOut)
{
    const int b   = blockIdx.x;
    const int tid = threadIdx.x;
    const int ln  = tid & 31;
    const int wv  = tid >> 5;
    const size_t base = (size_t)b * 3 * NPTS;
    const float* A0 = pc + base;
    const float* O0 = outp + base;

    __shared__ float sG[64];
    __shared__ __align__(16) float tile[8 * TROWS * TPITCH];
    if (tid < 64) sG[tid] = 0.f;
    // Zero entire staging tile once; loop rewrites only rows 0..6 per wave, so
    // rows 7..15 (the WMMA zero rows) remain zero -> unconditional operand reads.
    for (int i = tid; i < 8 * TROWS * TPITCH; i += 256) tile[i] = 0.f;

    float R1[9], t1[3], itv = 0.f;
    if constexpr (PASS2) {
        const float* rt = RT1 + b * 16;
        #pragma unroll
        for (int i = 0; i < 9; ++i) R1[i] = rt[i];
        t1[0] = rt[9]; t1[1] = rt[10]; t1[2] = rt[11];
        itv = interval[b];
    }
    __syncthreads();

    v8f acc = {};
    const int r    = ln & 15;
    const int half = ln >> 4;

    for (int it = 0; it < NPTS / 256; ++it) {
        const int p = it * 256 + tid;      // wave wv's lanes -> contiguous 32-pt window
        float ax = A0[p], ay = A0[NPTS + p], az = A0[2 * NPTS + p];
        float ox = O0[p], oy = O0[NPTS + p], oz = O0[2 * NPTS + p];
        if (it + 1 < NPTS / 256) {         // stream-prefetch next tile (global_prefetch_b8)
            __builtin_prefetch(&A0[p + 256], 0, 0);
            __builtin_prefetch(&A0[NPTS + p + 256], 0, 0);
            __builtin_prefetch(&A0[2 * NPTS + p + 256], 0, 0);
            __builtin_prefetch(&O0[p + 256], 0, 0);
            __builtin_prefetch(&O0[NPTS + p + 256], 0, 0);
            __builtin_prefetch(&O0[2 * NPTS + p + 256], 0, 0);
        }
        float bx = ax + ox, by = ay + oy, bz = az + oz;
        float w = 1.f;
        if constexpr (PASS2) {
            // sf_rg = R1*A + t1 - A ; proj = <sf_rg,A>/|A| ; mask = |(v*dt-proj)/v| < thr
            float sx = R1[0]*ax + R1[1]*ay + R1[2]*az + t1[0] - ax;
            float sy = R1[3]*ax + R1[4]*ay + R1[5]*az + t1[1] - ay;
            float sz = R1[6]*ax + R1[7]*ay + R1[8]*az + t1[2] - az;
            float nrm   = sqrtf(ax*ax + ay*ay + az*az);
            float proj  = (sx*ax + sy*ay + sz*az) / nrm;
            float vel   = feat[base + p];              // feature1[b][0][p]
            float resid = vel * itv - proj;
            w = (fabsf(resid / vel) < RIGID_THRES) ? 1.f : 0.f;
            maskOut[(size_t)b * NPTS + p] = w;
        }
        tile[TIDX(wv, 0, ln)] = w * ax;
        tile[TIDX(wv, 1, ln)] = w * ay;
        tile[TIDX(wv, 2, ln)] = w * az;
        tile[TIDX(wv, 3, ln)] = w * bx;
        tile[TIDX(wv, 4, ln)] = w * by;
        tile[TIDX(wv, 5, ln)] = w * bz;
        tile[TIDX(wv, 6, ln)] = w;
        __syncthreads();
        #pragma unroll
        for (int c = 0; c < 8; ++c) {
            const int p0 = c * 4 + half * 2;
            // lanes 0-15: row r, points (p0,p0+1) ; lanes 16-31: row r, (p0+2,p0+3)
            // = exact VGPR layout of both the 16x4 A-chunk and 4x16 B-chunk.
            v2f a = *(const v2f*)&tile[TIDX(wv, r, p0)];   // one ds_load_b64
            acc = __builtin_amdgcn_wmma_f32_16x16x4_f32(
                false, a, false, a, (short)0, acc, false, false);
        }
        __syncthreads();
    }

    // C layout: VGPR rr, lanes 0..15 hold G[rr][lane] (rows 8..15 are zero rows)
    #pragma unroll
    for (int rr = 0; rr < 8; ++rr) {
        float v = acc[rr];
        if (ln < 8) atomicAdd(&sG[rr * 8 + ln], v);
    }
    __syncthreads();
    if (tid < 64) G[(size_t)b * 64 + tid] = sG[tid];
}

// ---------------------------------------------------------------------------
// 3x3 Kabsch from Gram sums (one thread per batch)
// ---------------------------------------------------------------------------
__device__ inline void build_H(const float* __restrict__ g, float H[3][3],
                               float cA[3], float cB[3], float& Sw)
{
    const float invN = 1.f / (float)NPTS;
    float SA[3], SB[3];
    #pragma unroll
    for (int i = 0; i < 3; ++i) { SA[i] = g[i * 8 + 6]; SB[i] = g[(3 + i) * 8 + 6]; }
    Sw = g[6 * 8 + 6];
    #pragma unroll
    for (int i = 0; i < 3; ++i) { cA[i] = SA[i] * invN; cB[i] = SB[i] * invN; }
    #pragma unroll
    for (int i = 0; i < 3; ++i)
        #pragma unroll
        for (int j = 0; j < 3; ++j)
            H[i][j] = g[i * 8 + 3 + j] - cA[i] * SB[j] - SA[i] * cB[j]
                      + cA[i] * cB[j] * Sw;
}

__device__ inline void kabsch3x3(const float H[3][3], const float cA[3],
                                 const float cB[3], float R[3][3], float t[3])
{
    // Jacobi eigendecomposition of M = H^T H  ->  V, lambda
    float M[3][3], V[3][3] = {{1,0,0},{0,1,0},{0,0,1}};
    #pragma unroll
    for (int i = 0; i < 3; ++i)
        #pragma unroll
        for (int j = 0; j < 3; ++j)
            M[i][j] = H[0][i]*H[0][j] + H[1][i]*H[1][j] + H[2][i]*H[2][j];

    const int PQ[3][2] = {{0,1},{0,2},{1,2}};
    for (int sweep = 0; sweep < 8; ++sweep) {
        #pragma unroll
        for (int e = 0; e < 3; ++e) {
            const int p = PQ[e][0], q = PQ[e][1];
            float apq = M[p][q];
            if (fabsf(apq) > 1e-30f) {
                float tau = (M[q][q] - M[p][p]) / (2.f * apq);
                float tt  = ((tau >= 0.f) ? 1.f : -1.f)
                            / (fabsf(tau) + sqrtf(1.f + tau * tau));
                float c = 1.f / sqrtf(1.f + tt * tt);
                float s = tt * c;
                #pragma unroll
                for (int k = 0; k < 3; ++k) {      // M = M J (columns p,q)
                    float mkp = M[k][p], mkq = M[k][q];
                    M[k][p] = c * mkp - s * mkq;
                    M[k][q] = s * mkp + c * mkq;
                }
                #pragma unroll
                for (int k = 0; k < 3; ++k) {      // M = J^T M (rows p,q)
                    float mpk = M[p][k], mqk = M[q][k];
                    M[p][k] = c * mpk - s * mqk;
                    M[q][k] = s * mpk + c * mqk;
                }
                #pragma unroll
                for (int k = 0; k < 3; ++k) {      // V = V J
                    float vkp = V[k][p], vkq = V[k][q];
                    V[k][p] = c * vkp - s * vkq;
                    V[k][q] = s * vkp + c * vkq;
                }
            }
        }
    }
    // U columns = H v_i / sigma_i ; Z = V U^T (invariant to sign/order ambiguity)
    float U[3][3];
    #pragma unroll
    for (int i = 0; i < 3; ++i) {
        float lam = fmaxf(M[i][i], 0.f);
        float inv = 1.f / fmaxf(sqrtf(lam), 1e-20f);
        #pragma unroll
        for (int rr = 0; rr < 3; ++rr)
            U[rr][i] = (H[rr][0]*V[0][i] + H[rr][1]*V[1][i] + H[rr][2]*V[2][i]) * inv;
    }
    float Z[3][3];
    #pragma unroll
    for (int j = 0; j < 3; ++j)
        #pragma unroll
        for (int k = 0; k < 3; ++k)
            Z[j][k] = V[j][0]*U[k][0] + V[j][1]*U[k][1] + V[j][2]*U[k][2];

    float det = Z[0][0]*(Z[1][1]*Z[2][2] - Z[1][2]*Z[2][1])
              - Z[0][1]*(Z[1][0]*Z[2][2] - Z[1][2]*Z[2][0])
              + Z[0][2]*(Z[1][0]*Z[2][1] - Z[1][1]*Z[2][0]);
    float sgn = (det < 0.f) ? -1.f : 1.f;
    #pragma unroll
    for (int j = 0; j < 3; ++j)
        #pragma unroll
        for (int k = 0; k < 3; ++k)
            R[j][k] = ((j == 2) ? sgn : 1.f) * Z[j][k];   // reference flips V row 2
    #pragma unroll
    for (int i = 0; i < 3; ++i)
        t[i] = cB[i] - (R[i][0]*cA[0] + R[i][1]*cA[1] + R[i][2]*cA[2]);
}

__global__ void solve_pass1(const float* __restrict__ G, float* __restrict__ RT1)
{
    int b = blockIdx.x * blockDim.x + threadIdx.x;
    if (b >= BATCH) return;
    float H[3][3], cA[3], cB[3], Sw, R[3][3], t[3];
    build_H(G + (size_t)b * 64, H, cA, cB, Sw);
    kabsch3x3(H, cA, cB, R, t);
    float* o = RT1 + b * 16;
    #pragma unroll
    for (int i = 0; i < 3; ++i)
        #pragma unroll
        for (int j = 0; j < 3; ++j) o[i * 3 + j] = R[i][j];
    o[9] = t[0]; o[10] = t[1]; o[11] = t[2];
}

__global__ void solve_pass2(const float* __restrict__ G2, const float* __restrict__ RT1,
                            float* __restrict__ RT2, float* __restrict__ outPT)
{
    int b = blockIdx.x * blockDim.x + threadIdx.x;
    if (b >= BATCH) return;
    float H[3][3], cA[3], cB[3], Sw, R[3][3], t[3];
    build_H(G2 + (size_t)b * 64, H, cA, cB, Sw);
    kabsch3x3(H, cA, cB, R, t);
    bool use_rigid = (Sw * (1.f / (float)NPTS)) > RIGID_PCS;

    const float* r1 = RT1 + b * 16;
    float* pt = outPT + b * 16;           // pre_trans [4][4]
    #pragma unroll
    for (int i = 0; i < 3; ++i) {
        #pragma unroll
        for (int j = 0; j < 3; ++j)
            pt[i * 4 + j] = use_rigid ? R[i][j] : r1[i * 3 + j];
        pt[i * 4 + 3] = use_rigid ? t[i] : r1[9 + i];
    }
    pt[12] = 0.f; pt[13] = 0.f; pt[14] = 0.f; pt[15] = 1.f;

    float* o = RT2 + b * 16;              // trans_m + use_rigid flag for finalize
    #pragma unroll
    for (int i = 0; i < 3; ++i)
        #pragma unroll
        for (int j = 0; j < 3; ++j) o[i * 3 + j] = R[i][j];
    o[9] = t[0]; o[10] = t[1]; o[11] = t[2];
    o[12] = use_rigid ? 1.f : 0.f;
}

// sf_agg = (use_rigid && mask) ? R2*pc + t2 - pc : output
__global__ __launch_bounds__(256) void finalize(
    const float* __restrict__ pc, const float* __restrict__ outp,
    const float* __restrict__ RT2, const float* __restrict__ mask,
    float* __restrict__ sf)
{
    size_t idx = (size_t)blockIdx.x * blockDim.x + threadIdx.x;   // over B*N
    int b = (int)(idx / NPTS);
    int p = (int)(idx % NPTS);
    const float* rt = RT2 + (size_t)b * 16;
    bool rigid = rt[12] > 0.5f;
    float m = mask[idx];
    size_t base = (size_t)b * 3 * NPTS;
    float sx = outp[base + p], sy = outp[base + NPTS + p], sz = outp[base + 2 * NPTS + p];
    if (rigid && m > 0.5f) {
        float ax = pc[base + p], ay = pc[base + NPTS + p], az = pc[base + 2 * NPTS + p];
        sx = rt[0]*ax + rt[1]*ay + rt[2]*az + rt[9]  - ax;
        sy = rt[3]*ax + rt[4]*ay + rt[5]*az + rt[10] - ay;
        sz = rt[6]*ax + rt[7]*ay + rt[8]*az + rt[11] - az;
    }
    sf[base + p] = sx; sf[base + NPTS + p] = sy; sf[base + 2 * NPTS + p] = sz;
}

extern "C" void kernel_launch(void* const* d_in, const int* in_sizes, int n_in,
                              void* d_out, int out_size, void* d_ws, size_t ws_size,
                              hipStream_t stream)
{
    const float* pc1      = (const float*)d_in[0];   // [B,3,N]
    const float* feature1 = (const float*)d_in[1];   // [B,3,N]
    const float* outp     = (const float*)d_in[2];   // [B,3,N]
    const float* interval = (const float*)d_in[3];   // [B]

    float* out = (float*)d_out;
    float* sf  = out;                                       // [B,3,N]
    float* pt  = out + (size_t)BATCH * 3 * NPTS;            // [B,4,4]
    float* mk  = pt + (size_t)BATCH * 16;                   // [B,N] mask (0/1)

    // workspace: G1 | G2 | RT1 | RT2  = B*(64+64+16+16) floats = 640 KB
    float* ws  = (float*)d_ws;
    float* G1  = ws;
    float* G2  = G1 + (size_t)BATCH * 64;
    float* RT1 = G2 + (size_t)BATCH * 64;
    float* RT2 = RT1 + (size_t)BATCH * 16;

    gram_reduce<false><<<BATCH, 256, 0, stream>>>(pc1, outp, nullptr, nullptr,
                                                  nullptr, G1, nullptr);
    solve_pass1<<<BATCH / 256, 256, 0, stream>>>(G1, RT1);
    gram_reduce<true><<<BATCH, 256, 0, stream>>>(pc1, outp, feature1, interval,
                                                 RT1, G2, mk);
    solve_pass2<<<BATCH / 256, 256, 0, stream>>>(G2, RT1, RT2, pt);
    finalize<<<(BATCH * NPTS) / 256, 256, 0, stream>>>(pc1, outp, RT2, mk, sf);
}